// BalNoisedTopK_51642686767234
// MI455X (gfx1250) — compile-verified
//
#include <hip/hip_runtime.h>

#define D_DIM 32000
#define B_DIM 1024
#define NS 5
#define KP1 6
#define TPB 320                       // 10 waves of 32 (wave32 on gfx1250)
#define NWAVES (TPB / 32)
#define GROUPS (D_DIM / 4)            // 8000 groups of 4 consecutive d
#define ITERS (GROUPS / TPB)          // 25 exact

// Branchless insert of v into descending sorted top-6 list using v_med3_f32:
// 1 max + 5 med3 = 6 VALU ops (vs 12 for a min/max bubble pass).
__device__ __forceinline__ void insert6(float t[KP1], float v) {
    float n5 = __builtin_amdgcn_fmed3f(v, t[4], t[5]);
    float n4 = __builtin_amdgcn_fmed3f(v, t[3], t[4]);
    float n3 = __builtin_amdgcn_fmed3f(v, t[2], t[3]);
    float n2 = __builtin_amdgcn_fmed3f(v, t[1], t[2]);
    float n1 = __builtin_amdgcn_fmed3f(v, t[0], t[1]);
    float n0 = fmaxf(t[0], v);
    t[0] = n0; t[1] = n1; t[2] = n2; t[3] = n3; t[4] = n4; t[5] = n5;
}

__global__ __launch_bounds__(TPB)
void topk_row_kernel(const float* __restrict__ s,
                     const int* __restrict__ y,
                     const float* __restrict__ Z,
                     float* __restrict__ rownum) {
    const int b   = blockIdx.x;
    const int tid = threadIdx.x;
    const float* __restrict__ srow = s + (size_t)b * D_DIM;
    const float* __restrict__ zrow = Z + (size_t)b * D_DIM * NS;

    // 5 independent top-6 lists, descending, in VGPRs.
    float t[NS][KP1];
#pragma unroll
    for (int j = 0; j < NS; ++j)
#pragma unroll
        for (int i = 0; i < KP1; ++i) t[j][i] = -__builtin_inff();

    for (int it = 0; it < ITERS; ++it) {
        const int d0 = (tid + it * TPB) * 4;           // d0 % 4 == 0 -> 16B aligned

        // Prefetch next iteration's Z stream (emits global_prefetch_b8).
        __builtin_prefetch(zrow + (size_t)(d0 + 4 * TPB) * NS, 0, 1);

        const float4 sv = *reinterpret_cast<const float4*>(srow + d0);
        const float4* zq = reinterpret_cast<const float4*>(zrow + (size_t)d0 * NS);
        float4 q0 = zq[0], q1 = zq[1], q2 = zq[2], q3 = zq[3], q4 = zq[4];

        float zz[20];
        reinterpret_cast<float4*>(zz)[0] = q0;
        reinterpret_cast<float4*>(zz)[1] = q1;
        reinterpret_cast<float4*>(zz)[2] = q2;
        reinterpret_cast<float4*>(zz)[3] = q3;
        reinterpret_cast<float4*>(zz)[4] = q4;
        float ss[4] = { sv.x, sv.y, sv.z, sv.w };

#pragma unroll
        for (int dd = 0; dd < 4; ++dd) {
#pragma unroll
            for (int j = 0; j < NS; ++j) {
                insert6(t[j], ss[dd] + zz[dd * NS + j]);   // EPSILON == 1.0
            }
        }
    }

    // ---- intra-wave merge (wave32): butterfly over 32 lanes ----
#pragma unroll
    for (int off = 16; off > 0; off >>= 1) {
#pragma unroll
        for (int j = 0; j < NS; ++j) {
            float p[KP1];
#pragma unroll
            for (int i = 0; i < KP1; ++i) p[i] = __shfl_xor(t[j][i], off, 32);
#pragma unroll
            for (int i = 0; i < KP1; ++i) insert6(t[j], p[i]);
        }
    }

    // ---- cross-wave merge via LDS ----
    __shared__ float red[NWAVES][NS][KP1];
    __shared__ float kth[NS];
    const int wave = tid >> 5;
    const int lane = tid & 31;
    if (lane == 0) {
#pragma unroll
        for (int j = 0; j < NS; ++j)
#pragma unroll
            for (int i = 0; i < KP1; ++i) red[wave][j][i] = t[j][i];
    }
    __syncthreads();

    if (tid < NS) {
        const int j = tid;
        float m[KP1];
#pragma unroll
        for (int i = 0; i < KP1; ++i) m[i] = red[0][j][i];
#pragma unroll
        for (int w = 1; w < NWAVES; ++w)
#pragma unroll
            for (int i = 0; i < KP1; ++i) insert6(m, red[w][j][i]);
        kth[j] = m[KP1 - 1];                 // 6th-largest for this sample
    }
    __syncthreads();

    if (tid == 0) {
        const float skp1 = (kth[0] + kth[1] + kth[2] + kth[3] + kth[4]) * 0.2f;
        const float cs   = srow[y[b]];
        rownum[b] = fmaxf(0.0f, 1.0f + skp1 - cs);
    }
}

// Deterministic fixed-order mean over B rows (no float atomics).
__global__ __launch_bounds__(256)
void mean_kernel(const float* __restrict__ rownum, float* __restrict__ out) {
    __shared__ float acc[256];
    const int tid = threadIdx.x;
    float v = 0.0f;
    for (int i = tid; i < B_DIM; i += 256) v += rownum[i];
    acc[tid] = v;
    __syncthreads();
#pragma unroll
    for (int off = 128; off > 0; off >>= 1) {
        if (tid < off) acc[tid] += acc[tid + off];
        __syncthreads();
    }
    if (tid == 0) out[0] = acc[0] * (1.0f / (float)B_DIM);
}

extern "C" void kernel_launch(void* const* d_in, const int* in_sizes, int n_in,
                              void* d_out, int out_size, void* d_ws, size_t ws_size,
                              hipStream_t stream) {
    (void)in_sizes; (void)n_in; (void)out_size; (void)ws_size;
    const float* s = (const float*)d_in[0];
    const int*   y = (const int*)d_in[1];
    const float* Z = (const float*)d_in[2];
    float* rownum  = (float*)d_ws;            // B_DIM floats of scratch
    float* out     = (float*)d_out;

    topk_row_kernel<<<B_DIM, TPB, 0, stream>>>(s, y, Z, rownum);
    mean_kernel<<<1, 256, 0, stream>>>(rownum, out);
}